// DynamicUpsamplingFilter_283467842094
// MI455X (gfx1250) — compile-verified
//
#include <hip/hip_runtime.h>
#include <hip/hip_bf16.h>

// Dynamic Upsampling Filter for MI455X (gfx1250, wave32).
//
// Shapes (fixed by setup_inputs):
//   x:       (4, 3, 180, 320)           f32
//   filters: (4, 25, 16, 180, 320)      f32
//   out:     (4, 48, 180, 320)          f32, out[n, c*16+u, h, w]
//
// Memory-bound kernel (~1.3 FLOP/byte): filters (369 MB) are streamed once,
// coalesced along w, through LDS into V_WMMA_F32_16X16X4_F32 fragments.
// One wave computes one pixel's (3x25)@(25x16) matmul as 7 chained K=4
// f32 WMMAs (bit-exact f32 accumulation, no precision loss vs reference).

typedef __attribute__((ext_vector_type(2))) float v2f;
typedef __attribute__((ext_vector_type(8))) float v8f;

#define N_     4
#define C_     3
#define H_     180
#define W_     320
#define HW_    (H_ * W_)          // 57600
#define P_     25                 // 5x5 taps
#define U_     16                 // upsampling_square
#define PU_    (P_ * U_)          // 400 filter values per pixel
#define TILE_  32                 // pixels (w) per workgroup
#define FS_    401                // LDS per-pixel filter stride (odd -> no bank conflicts)
#define XCOLS_ 36                 // 32 + 4 halo columns
#define OSTR_  33                 // out staging stride (odd)

__global__ void __launch_bounds__(128)
duf_wmma_kernel(const float* __restrict__ x,
                const float* __restrict__ filt,
                float* __restrict__ out) {
  __shared__ float filtS[TILE_ * FS_];      // 51328 B, [pixel][p*16+u]
  __shared__ float xS[C_ * 5 * XCOLS_];     //  2160 B, [c][row 0..4][col 0..35]
  __shared__ float outS[48 * OSTR_];        //  6336 B, [c*16+u][pixel]

  const int tid  = threadIdx.x;             // 128 threads = 4 wave32
  const int lane = tid & 31;
  const int wid  = tid >> 5;

  const int w0 = blockIdx.x * TILE_;
  const int h  = blockIdx.y;
  const int n  = blockIdx.z;

  // ---- Phase 1a: stream per-pixel 25x16 filter matrices, coalesced along w.
  // Per wave iteration: lanes = 32 consecutive w pixels, one (p,u) slot.
  {
    const float* fbase = filt + (size_t)n * PU_ * HW_ + (size_t)h * W_ + w0;
    for (int idx = tid; idx < TILE_ * PU_; idx += 128) {
      const int q = idx & (TILE_ - 1);      // pixel within tile (lane-contiguous)
      const int t = idx >> 5;               // t = p*16 + u
      filtS[q * FS_ + t] = fbase[(size_t)t * HW_ + q];  // odd stride: conflict-free
    }
  }

  // ---- Phase 1b: stage x halo tile (3ch x 5 rows x 36 cols), zero-padded borders.
  for (int idx = tid; idx < C_ * 5 * XCOLS_; idx += 128) {
    const int c   = idx / (5 * XCOLS_);
    const int rem = idx - c * (5 * XCOLS_);
    const int r   = rem / XCOLS_;
    const int col = rem - r * XCOLS_;
    const int gh  = h - 2 + r;
    const int gw  = w0 - 2 + col;
    float v = 0.0f;
    if (gh >= 0 && gh < H_ && gw >= 0 && gw < W_)
      v = x[((size_t)(n * C_ + c) * H_ + gh) * W_ + gw];
    xS[idx] = v;
  }
  __syncthreads();

  // ---- Phase 2: per-pixel matmul on the WMMA pipe. 8 pixels per wave.
  // f32 16x16x4 layouts (wave32):
  //   A 16x4 : lane L holds row M=L%16; VGPR0 = K(2*(L/16)),   VGPR1 = K(+1)
  //   B  4x16: lane L holds col N=L%16; VGPR0 = row K(2*(L/16)), VGPR1 = row K(+1)
  //   C 16x16: acc[r] at lanes 0-15 is row M=r, col N=lane.
  // Row m of D depends only on row m of A/C, so garbage rows M>=3 are harmless;
  // K=25..27 tail is handled by zeroing B (0 * finite-A = 0).
  const int u  = lane & 15;                 // B column / output u / A row (=channel)
  const int hi = lane >> 4;                 // K sub-slot within a 4-wide call
  const int c2 = (u < C_) ? u : 0;          // clamp channel for padding rows

  for (int pq = 0; pq < TILE_ / 4; ++pq) {
    const int q  = wid * (TILE_ / 4) + pq;
    const int qb = q * FS_;
    v8f acc = {0.f, 0.f, 0.f, 0.f, 0.f, 0.f, 0.f, 0.f};
#pragma unroll
    for (int t = 0; t < 7; ++t) {
      const int k0 = 4 * t + 2 * hi;
      const int k1 = k0 + 1;
      const int p0 = (k0 < P_) ? k0 : (P_ - 1);   // finite A for padded K
      const int p1 = (k1 < P_) ? k1 : (P_ - 1);
      v2f a, b;
      // A: patch value of channel c2 at tap p = i*5+j -> x[h-2+i][w0+q-2+j]
      a.x = xS[(c2 * 5 + p0 / 5) * XCOLS_ + q + (p0 % 5)];
      a.y = xS[(c2 * 5 + p1 / 5) * XCOLS_ + q + (p1 % 5)];
      // B: filt[p][u] of this pixel; lanes 0-15 banks u+0..15, lanes 16-31
      // banks +32 -> conflict-free within the read.
      b.x = (k0 < P_) ? filtS[qb + k0 * U_ + u] : 0.0f;
      b.y = (k1 < P_) ? filtS[qb + k1 * U_ + u] : 0.0f;
      acc = __builtin_amdgcn_wmma_f32_16x16x4_f32(
          /*neg_a=*/false, a, /*neg_b=*/false, b,
          /*c_mod=*/(short)0, acc, /*reuse_a=*/false, /*reuse_b=*/false);
    }
    // Extract rows 0..2 (channels) from C: acc[c] at lanes 0-15, N=u.
    if (lane < 16) {
      outS[(0 * U_ + u) * OSTR_ + q] = acc[0];
      outS[(1 * U_ + u) * OSTR_ + q] = acc[1];
      outS[(2 * U_ + u) * OSTR_ + q] = acc[2];
    }
  }
  __syncthreads();

  // ---- Phase 3: coalesced stores, lanes along contiguous w.
  {
    float* obase = out + (size_t)n * 48 * HW_ + (size_t)h * W_ + w0;
    for (int idx = tid; idx < 48 * TILE_; idx += 128) {
      const int q  = idx & (TILE_ - 1);
      const int ou = idx >> 5;              // c*16 + u
      obase[(size_t)ou * HW_ + q] = outS[ou * OSTR_ + q];
    }
  }
}

extern "C" void kernel_launch(void* const* d_in, const int* in_sizes, int n_in,
                              void* d_out, int out_size, void* d_ws, size_t ws_size,
                              hipStream_t stream) {
  (void)in_sizes; (void)n_in; (void)out_size; (void)d_ws; (void)ws_size;
  const float* x = (const float*)d_in[0];        // (4,3,180,320)
  const float* f = (const float*)d_in[1];        // (4,25,16,180,320)
  float* out = (float*)d_out;                    // (4,48,180,320)
  dim3 grid(W_ / TILE_, H_, N_);                 // (10, 180, 4)
  duf_wmma_kernel<<<grid, dim3(128, 1, 1), 0, stream>>>(x, f, out);
}